// QuaternionSpectralLayer_24352464570152
// MI455X (gfx1250) — compile-verified
//
#include <hip/hip_runtime.h>

#define GRID_H 64
#define GRID_W 64
#define FREQ_W 33
#define NFREQ  (GRID_H * FREQ_W)   /* 2112 */
#define TPITCH 65                  /* padded LDS tile pitch (bank-conflict-free) */

typedef __attribute__((ext_vector_type(2))) float v2f;
typedef __attribute__((ext_vector_type(8))) float v8f;

// ---------------------------------------------------------------------------
// 64-point complex FFT on a strided LDS array, serial per thread (radix-2 DIT).
// Twiddles from a 32-entry LDS LUT: tw[m] = exp(-2*pi*i*m/64).
// ---------------------------------------------------------------------------
__device__ __forceinline__ void fft64_strided(float2* d, int stride, bool inverse,
                                              const float2* __restrict__ tw) {
  for (int i = 0; i < 64; ++i) {
    int j = (int)(__brev((unsigned)i) >> 26);
    if (j > i) {
      float2 t = d[i * stride];
      d[i * stride] = d[j * stride];
      d[j * stride] = t;
    }
  }
  const float sf = inverse ? -1.0f : 1.0f;
  for (int len = 2, step = 32; len <= 64; len <<= 1, step >>= 1) {
    for (int i = 0; i < 64; i += len) {
      for (int j = 0; j < (len >> 1); ++j) {
        float2 w = tw[j * step];
        float c = w.x, s = sf * w.y;
        float2 u = d[(i + j) * stride];
        float2 v = d[(i + j + (len >> 1)) * stride];
        float vr = v.x * c - v.y * s;
        float vi = v.x * s + v.y * c;
        d[(i + j) * stride]              = make_float2(u.x + vr, u.y + vi);
        d[(i + j + (len >> 1)) * stride] = make_float2(u.x - vr, u.y - vi);
      }
    }
  }
}

__device__ __forceinline__ void init_twiddles(float2* twsh, int t) {
  if (t < 32) {
    float s, c;
    __sincosf(-6.283185307179586f * (float)t * (1.0f / 64.0f), &s, &c);
    twsh[t] = make_float2(c, s);
  }
}

// ---------------------------------------------------------------------------
// Pack kernels into frequency-major SoA layout, W transposed to [o][i]:
//   Wpack[f*8192 + (p*2+reim)*1024 + o*32 + i]
// ---------------------------------------------------------------------------
__global__ void pack_w_kernel(const float* __restrict__ kr,
                              const float* __restrict__ ki,
                              float* __restrict__ Wpack, int p) {
  int io = blockIdx.x;                         // 0..1023 == i*32+o
  int oi = ((io & 31) << 5) | (io >> 5);       // o*32+i
  for (int f = threadIdx.x; f < NFREQ; f += blockDim.x) {
    size_t base = (size_t)f * 8192 + (size_t)(p * 2) * 1024 + oi;
    Wpack[base]        = kr[(size_t)io * NFREQ + f];
    Wpack[base + 1024] = ki[(size_t)io * NFREQ + f];
  }
}

// ---------------------------------------------------------------------------
// Forward rfft2 of one 64x64 image -> frequency-major SoA
//   Xpack[f*8192 + (q*2+reim)*1024 + b*32 + i]
// ---------------------------------------------------------------------------
__global__ void fft_fwd_kernel(const float* __restrict__ x,
                               float* __restrict__ Xpack) {
  __shared__ float2 tile[64 * TPITCH];
  __shared__ float2 twsh[32];
  int img = blockIdx.x;                 // b*128 + (p*32 + i)
  int b = img >> 7;
  int c = img & 127;
  int p = c >> 5;
  int i = c & 31;
  const float* src = x + (size_t)img * 4096;
  int t = threadIdx.x;                  // 0..63

  init_twiddles(twsh, t);
  for (int idx = t; idx < 4096; idx += 64)
    tile[(idx >> 6) * TPITCH + (idx & 63)] = make_float2(src[idx], 0.0f);
  __syncthreads();

  fft64_strided(&tile[t * TPITCH], 1, false, twsh);            // row FFT (W)
  __syncthreads();
  if (t < FREQ_W) fft64_strided(&tile[t], TPITCH, false, twsh); // col FFT (H)
  __syncthreads();

  for (int wf = 0; wf < FREQ_W; ++wf) {
    int f = t * FREQ_W + wf;
    size_t base = (size_t)f * 8192 + (size_t)(p * 2) * 1024 + b * 32 + i;
    float2 v = tile[t * TPITCH + wf];
    Xpack[base]        = v.x;
    Xpack[base + 1024] = v.y;
  }
}

// ---------------------------------------------------------------------------
// Per-frequency quaternion complex contraction with WMMA f32 16x16x4.
// 256 threads = 8 waves; wave w: p_out = w>>1, bTile = w&1.
// q = p ^ p_out; Hamilton sign nibble table packed in 0x428E.
// ---------------------------------------------------------------------------
__global__ void __launch_bounds__(256)
qgemm_kernel(const float* __restrict__ Xpack,
             const float* __restrict__ Wpack,
             float2* __restrict__ Pimg) {
  __shared__ float Xs[8192];   // 32 KB : [q][reim][b*32+i]
  __shared__ float Ws[8192];   // 32 KB : [p][reim][o*32+i]
  int f    = blockIdx.x;
  int tid  = threadIdx.x;
  int wv   = tid >> 5;         // 0..7
  int pout = wv >> 1;
  int bT   = wv & 1;
  int lane = tid & 31;
  int row  = lane & 15;
  int hi   = lane >> 4;

  {
    const float4* gx = (const float4*)(Xpack + (size_t)f * 8192);
    const float4* gw = (const float4*)(Wpack + (size_t)f * 8192);
    float4* sx = (float4*)Xs;
    float4* sw = (float4*)Ws;
    for (int idx = tid; idx < 2048; idx += 256) {
      sx[idx] = gx[idx];
      sw[idx] = gw[idx];
    }
  }
  __syncthreads();

  v8f zero = {0.f, 0.f, 0.f, 0.f, 0.f, 0.f, 0.f, 0.f};
  v8f accRp[2] = {zero, zero};
  v8f accRm[2] = {zero, zero};
  v8f accIp[2] = {zero, zero};
  v8f accIm[2] = {zero, zero};

  const int arow = (bT * 16 + row) * 32;   // A row base (floats within plane)

// One real (16..32)x32x32 matmul-accumulate:
//   dst[oT] += X_q(plane ax)[bT rows] * W_p(plane bx)
#define MM(dst, qq, ax, pp, bx)                                                \
  do {                                                                         \
    const float* xpl = Xs + ((qq) * 2 + (ax)) * 1024 + arow;                   \
    const float* wpl = Ws + ((pp) * 2 + (bx)) * 1024 + row * 32;               \
    _Pragma("unroll 2")                                                        \
    for (int k = 0; k < 8; ++k) {                                              \
      int ib = k * 4 + 2 * hi;                                                 \
      v2f a = *(const v2f*)(xpl + ib);                                         \
      _Pragma("unroll")                                                        \
      for (int oT = 0; oT < 2; ++oT) {                                         \
        v2f bv = *(const v2f*)(wpl + oT * 512 + ib);                           \
        dst[oT] = __builtin_amdgcn_wmma_f32_16x16x4_f32(                       \
            false, a, false, bv, (short)0, dst[oT], false, false);             \
      }                                                                        \
    }                                                                          \
  } while (0)

// Complex matmul-accumulate: Cr += XrWr - XiWi ; Ci += XrWi + XiWr
#define CPLX_MM(dRp, dRm, dI, qq, pp) \
  do {                                \
    MM(dRp, qq, 0, pp, 0);            \
    MM(dRm, qq, 1, pp, 1);            \
    MM(dI,  qq, 0, pp, 1);            \
    MM(dI,  qq, 1, pp, 0);            \
  } while (0)

#pragma unroll
  for (int j = 0; j < 4; ++j) {       // p = j
    int q = j ^ pout;
    bool neg = (0x428E >> (pout * 4 + j)) & 1;   // Hamilton sign for C[p][q]
    if (!neg) {
      CPLX_MM(accRp, accRm, accIp, q, j);
    } else {
      CPLX_MM(accRm, accRp, accIm, q, j);
    }
  }
#undef CPLX_MM
#undef MM

  // combine signs and scatter to image-major spectral output
#pragma unroll
  for (int oT = 0; oT < 2; ++oT) {
    v8f cr = accRp[oT] - accRm[oT];
    v8f ci = accIp[oT] - accIm[oT];
#pragma unroll
    for (int r = 0; r < 8; ++r) {
      int bb = bT * 16 + r + (hi ? 8 : 0);
      int oo = oT * 16 + row;
      size_t chan = (size_t)bb * 128 + pout * 32 + oo;
      Pimg[chan * NFREQ + f] = make_float2(cr[r], ci[r]);
    }
  }
}

// ---------------------------------------------------------------------------
// Inverse rfft2 of one spectral image -> real output image
// ---------------------------------------------------------------------------
__global__ void fft_inv_kernel(const float2* __restrict__ Pimg,
                               float* __restrict__ out) {
  __shared__ float2 tile[64 * TPITCH];
  __shared__ float2 twsh[32];
  int img = blockIdx.x;                 // b*128 + p*32 + o
  const float2* src = Pimg + (size_t)img * NFREQ;
  int t = threadIdx.x;

  init_twiddles(twsh, t);
  for (int idx = t; idx < NFREQ; idx += 64) {
    int h = idx / FREQ_W, wf = idx % FREQ_W;
    tile[h * TPITCH + wf] = src[idx];
  }
  __syncthreads();

  if (t < FREQ_W) fft64_strided(&tile[t], TPITCH, true, twsh);  // inv col FFT
  __syncthreads();

  for (int w = FREQ_W; w < 64; ++w) {                  // Hermitian expansion
    float2 v = tile[t * TPITCH + (64 - w)];
    tile[t * TPITCH + w] = make_float2(v.x, -v.y);
  }
  fft64_strided(&tile[t * TPITCH], 1, true, twsh);     // inverse row FFT
  __syncthreads();

  const float scale = 1.0f / 4096.0f;
  float* dst = out + (size_t)img * 4096;
  for (int idx = t; idx < 4096; idx += 64)
    dst[idx] = tile[(idx >> 6) * TPITCH + (idx & 63)].x * scale;
}

// ---------------------------------------------------------------------------
extern "C" void kernel_launch(void* const* d_in, const int* in_sizes, int n_in,
                              void* d_out, int out_size, void* d_ws, size_t ws_size,
                              hipStream_t stream) {
  (void)in_sizes; (void)n_in; (void)out_size; (void)ws_size;

  const float* x = (const float*)d_in[0];
  // d_in[1..4] = kr_w, kr_x, kr_y, kr_z ; d_in[5..8] = ki_w, ki_x, ki_y, ki_z

  const size_t SLAB_F = (size_t)8192 * NFREQ;      // floats per SoA buffer
  float*  Xpack = (float*)d_ws;
  float*  Wpack = Xpack + SLAB_F;
  float2* Pimg  = (float2*)(Wpack + SLAB_F);

  for (int p = 0; p < 4; ++p)
    pack_w_kernel<<<1024, 256, 0, stream>>>((const float*)d_in[1 + p],
                                            (const float*)d_in[5 + p], Wpack, p);
  fft_fwd_kernel<<<4096, 64, 0, stream>>>(x, Xpack);
  qgemm_kernel<<<NFREQ, 256, 0, stream>>>(Xpack, Wpack, Pimg);
  fft_inv_kernel<<<4096, 64, 0, stream>>>(Pimg, (float*)d_out);
}